// ConvAttentionHybrid_65481071398939
// MI455X (gfx1250) — compile-verified
//
#include <hip/hip_runtime.h>
#include <math.h>

typedef __attribute__((ext_vector_type(2))) float v2f;
typedef __attribute__((ext_vector_type(8))) float v8f;

#define S_TOTAL 16384          // 128*128 sequence positions
#define TILE    16
#define N_TILES (S_TOTAL / TILE)   // 1024 tiles of 16
#define WAVES_PER_BLOCK 8

// ---------------------------------------------------------------------------
// Kernel 1: 2x2 valid conv + bias + sigmoid -> f[16384] in workspace
// ---------------------------------------------------------------------------
__global__ __launch_bounds__(256) void conv_sigmoid_kernel(
    const float* __restrict__ data,   // 129x129
    const float* __restrict__ cw,     // 4 (kh-major: w00,w01,w10,w11)
    const float* __restrict__ cb,     // 1
    float* __restrict__ f)            // out: 16384
{
    int t = blockIdx.x * blockDim.x + threadIdx.x;
    if (t >= S_TOTAL) return;
    int i = t >> 7;        // row in 128x128 output
    int j = t & 127;       // col
    const float* r0 = data + i * 129 + j;
    const float* r1 = r0 + 129;
    float x = cb[0];
    x = fmaf(r0[0], cw[0], x);
    x = fmaf(r0[1], cw[1], x);
    x = fmaf(r1[0], cw[2], x);
    x = fmaf(r1[1], cw[3], x);
    // sigmoid
    f[t] = 1.0f / (1.0f + __expf(-x));
}

// ---------------------------------------------------------------------------
// Kernel 2: flash-style attention over rank-1 embeddings.
// Each wave owns one 16-row s-tile; streams all 1024 t-tiles.
// logits tile (16x16) via V_WMMA_F32_16X16X4_F32 (M=16,N=16,K=E=4).
// scale*log2(e) folded into Q fragment so softmax weights come straight
// from v_exp_f32 (exp2) with no per-element pre-multiply.
// Accumulates per-row Z = sum_t w and MZ = sum_t w*f[t]; writes partial
// sum_s (MZ/Z) per tile (deterministic, no atomics).
// ---------------------------------------------------------------------------
__global__ __launch_bounds__(256) void attn_kernel(
    const float* __restrict__ f_g,    // 16384 sigmoid outputs
    const float* __restrict__ Wq, const float* __restrict__ bq,
    const float* __restrict__ Wk, const float* __restrict__ bk,
    float* __restrict__ msum_part)    // out: 1024 partials
{
    __shared__ __align__(16) float fsh[S_TOTAL];   // 64 KB of 320 KB WGP LDS

    const int tid = threadIdx.x;

    // Async global->LDS stage of f (16 B / lane / issue, ASYNCcnt-tracked).
    // LDS address taken per lane from the vdst VGPR; 64-bit global address in vaddr.
    for (int base = 0; base < S_TOTAL; base += 256 * 4) {
        const int idx = base + tid * 4;
        unsigned lds_off = (unsigned)(size_t)(&fsh[idx]);
        const float* gptr = f_g + idx;
        asm volatile("global_load_async_to_lds_b128 %0, %1, off"
                     :
                     : "v"(lds_off), "v"(gptr)
                     : "memory");
    }
    asm volatile("s_wait_asynccnt 0x0" ::: "memory");
    __syncthreads();

    const int lane = tid & 31;
    const int wave = tid >> 5;
    const int tile = blockIdx.x * WAVES_PER_BLOCK + wave;   // 0..1023
    const int sbase = tile * TILE;

    // f32 16x4 A-fragment layout: lanes 0-15 -> M=lane, K=0/1 in v[0]/v[1];
    // lanes 16-31 -> M=lane-16, K=2/3.  B (4x16) mirrors with N instead of M.
    const int half = lane >> 4;       // 0: e=0,1   1: e=2,3
    const int col  = lane & 15;       // M (for A) / N (for B) index
    const int e0 = 2 * half;
    const int e1 = e0 + 1;

    // q[s,e] = f[s]*rowsum(Wq)[e] + bq[e];  k[t,e] = f[t]*rowsum(Wk)[e] + bk[e]
    const float aq0 = Wq[e0*4+0] + Wq[e0*4+1] + Wq[e0*4+2] + Wq[e0*4+3];
    const float aq1 = Wq[e1*4+0] + Wq[e1*4+1] + Wq[e1*4+2] + Wq[e1*4+3];
    const float ak0 = Wk[e0*4+0] + Wk[e0*4+1] + Wk[e0*4+2] + Wk[e0*4+3];
    const float ak1 = Wk[e1*4+0] + Wk[e1*4+1] + Wk[e1*4+2] + Wk[e1*4+3];
    const float bq0 = bq[e0], bq1 = bq[e1];
    const float bk0 = bk[e0], bk1 = bk[e1];

    // 1/sqrt(E) softmax scale and log2(e) (exp -> exp2) folded into Q
    const float sl2e = 0.5f * 1.44269504088896340736f;

    // Q fragment for this s-tile (constant over the t loop)
    const float fs = fsh[sbase + col];
    v2f afrag;
    afrag.x = sl2e * fmaf(fs, aq0, bq0);
    afrag.y = sl2e * fmaf(fs, aq1, bq1);

    // per-lane accumulators: row m = r + 8*half, column n = col
    float z[8], mz[8];
#pragma unroll
    for (int r = 0; r < 8; ++r) { z[r] = 0.0f; mz[r] = 0.0f; }

    for (int tt = 0; tt < N_TILES; ++tt) {
        const float ft = fsh[tt * TILE + col];
        v2f bfrag;
        bfrag.x = fmaf(ft, ak0, bk0);
        bfrag.y = fmaf(ft, ak1, bk1);

        v8f cacc = {};
        // D = (sl2e*Q) x K^T : 16x16 log2-scaled logits
        v8f d = __builtin_amdgcn_wmma_f32_16x16x4_f32(
            /*neg_a=*/false, afrag, /*neg_b=*/false, bfrag,
            /*c_mod=*/(short)0, cacc, /*reuse_a=*/false, /*reuse_b=*/false);

#pragma unroll
        for (int r = 0; r < 8; ++r) {
            const float p = __builtin_amdgcn_exp2f(d[r]);  // v_exp_f32 direct
            z[r]  += p;
            mz[r]  = fmaf(p, ft, mz[r]);
        }
    }

    // reduce over the 16 lanes (columns n) of each half-wave
#pragma unroll
    for (int mask = 1; mask <= 8; mask <<= 1) {
#pragma unroll
        for (int r = 0; r < 8; ++r) {
            z[r]  += __shfl_xor(z[r],  mask, 32);
            mz[r] += __shfl_xor(mz[r], mask, 32);
        }
    }

    // sum over this half's 8 rows of m_s = MZ/Z
    float msum = 0.0f;
#pragma unroll
    for (int r = 0; r < 8; ++r) msum += mz[r] / z[r];
    // combine the two halves (rows 0-7 and 8-15)
    msum += __shfl_xor(msum, 16, 32);

    if (lane == 0) msum_part[tile] = msum;
}

// ---------------------------------------------------------------------------
// Kernel 3: reduce partials; mean = (sum(Wv)*sum_s m_s + S*sum(bv)) / (S*E)
// ---------------------------------------------------------------------------
__global__ __launch_bounds__(256) void finalize_kernel(
    const float* __restrict__ msum_part,
    const float* __restrict__ Wv, const float* __restrict__ bv,
    float* __restrict__ out)
{
    __shared__ float red[256];
    const int tid = threadIdx.x;
    float s = 0.0f;
    for (int i = tid; i < N_TILES; i += 256) s += msum_part[i];
    red[tid] = s;
    __syncthreads();
    for (int off = 128; off > 0; off >>= 1) {
        if (tid < off) red[tid] += red[tid + off];
        __syncthreads();
    }
    if (tid == 0) {
        float sumC = 0.0f;
        for (int i = 0; i < 16; ++i) sumC += Wv[i];
        const float sumBv = bv[0] + bv[1] + bv[2] + bv[3];
        const float msum = red[0];
        out[0] = (sumC * msum + (float)S_TOTAL * sumBv)
                 / ((float)S_TOTAL * 4.0f);
    }
}

// ---------------------------------------------------------------------------
extern "C" void kernel_launch(void* const* d_in, const int* in_sizes, int n_in,
                              void* d_out, int out_size, void* d_ws, size_t ws_size,
                              hipStream_t stream) {
    const float* data = (const float*)d_in[0];   // 129*129
    const float* cw   = (const float*)d_in[1];   // 4
    const float* cb   = (const float*)d_in[2];   // 1
    const float* Wq   = (const float*)d_in[3];   // 16
    const float* bq   = (const float*)d_in[4];   // 4
    const float* Wk   = (const float*)d_in[5];   // 16
    const float* bk   = (const float*)d_in[6];   // 4
    const float* Wv   = (const float*)d_in[7];   // 16
    const float* bv   = (const float*)d_in[8];   // 4

    float* f_ws      = (float*)d_ws;             // 16384 floats
    float* msum_part = f_ws + S_TOTAL;           // 1024 floats

    conv_sigmoid_kernel<<<S_TOTAL / 256, 256, 0, stream>>>(data, cw, cb, f_ws);
    attn_kernel<<<N_TILES / WAVES_PER_BLOCK, 256, 0, stream>>>(
        f_ws, Wq, bq, Wk, bk, msum_part);
    finalize_kernel<<<1, 256, 0, stream>>>(msum_part, Wv, bv, (float*)d_out);
}